// Neural_network_43379169689954
// MI455X (gfx1250) — compile-verified
//
#include <hip/hip_runtime.h>
#include <stdint.h>

// ---------------------------------------------------------------------------
// Problem constants (reference: N=1e6 samples, K=64 components, D=16 dims)
// ---------------------------------------------------------------------------
#define KCOMP 64
#define DDIM  16

typedef __attribute__((ext_vector_type(2))) float v2f;
typedef __attribute__((ext_vector_type(8))) float v8f;

// ---------------------------------------------------------------------------
// Threefry-2x32 with key = (0, 42)  (jax.random.key(42))
// ---------------------------------------------------------------------------
struct TF2 { uint32_t a, b; };

__device__ __forceinline__ uint32_t rotl32(uint32_t x, int r) {
    return (x << r) | (x >> (32 - r));
}

__device__ __forceinline__ TF2 threefry2x32_key42(uint32_t x0, uint32_t x1) {
    const uint32_t ks0 = 0u;
    const uint32_t ks1 = 42u;
    const uint32_t ks2 = ks0 ^ ks1 ^ 0x1BD11BDAu;
    x0 += ks0; x1 += ks1;
#define TF_RND(r) { x0 += x1; x1 = rotl32(x1, r); x1 ^= x0; }
    TF_RND(13) TF_RND(15) TF_RND(26) TF_RND(6)   x0 += ks1; x1 += ks2 + 1u;
    TF_RND(17) TF_RND(29) TF_RND(16) TF_RND(24)  x0 += ks2; x1 += ks0 + 2u;
    TF_RND(13) TF_RND(15) TF_RND(26) TF_RND(6)   x0 += ks0; x1 += ks1 + 3u;
    TF_RND(17) TF_RND(29) TF_RND(16) TF_RND(24)  x0 += ks1; x1 += ks2 + 4u;
    TF_RND(13) TF_RND(15) TF_RND(26) TF_RND(6)   x0 += ks2; x1 += ks0 + 5u;
#undef TF_RND
    TF2 r; r.a = x0; r.b = x1; return r;
}

__device__ __forceinline__ float gumbel_from_bits(uint32_t bits) {
    // jax uniform: bitcast((bits>>9)|0x3f800000) - 1  in [0,1), clamp to tiny
    float u = __uint_as_float((bits >> 9) | 0x3f800000u) - 1.0f;
    u = fmaxf(u, 1.1754944e-38f);
    return -__logf(-__logf(u));
}

// ---------------------------------------------------------------------------
// Kernel 0: zero counters, compute log-probabilities  (1 block, 64 threads)
// ---------------------------------------------------------------------------
__global__ void k_init(const float* __restrict__ pi,
                       int* __restrict__ hist, int* __restrict__ fill,
                       float* __restrict__ logp) {
    __shared__ float sp[KCOMP];
    __shared__ float ssum;
    int t = threadIdx.x;
    hist[t] = 0;
    fill[t] = 0;
    sp[t] = pi[t];
    __syncthreads();
    if (t == 0) {
        float s = 0.f;
        for (int i = 0; i < KCOMP; ++i) s += sp[i];
        ssum = s;
    }
    __syncthreads();
    logp[t] = __logf(sp[t]) - __logf(ssum);
}

// ---------------------------------------------------------------------------
// Kernel 1: z[n] = argmax_k (logp[k] + gumbel(n,k)); shared-hash pairing:
// counters i and i+NK/2 come from one threefry call -> one thread produces
// z[n] and z[n + N/2]. Histogram staged in LDS.
// ---------------------------------------------------------------------------
__global__ __launch_bounds__(256)
void k_sample(const float* __restrict__ logp,
              int* __restrict__ z, int* __restrict__ hist,
              int H, uint32_t NK2) {
    __shared__ int   lhist[KCOMP];
    __shared__ float slogp[KCOMP];
    int t = threadIdx.x;
    if (t < KCOMP) { lhist[t] = 0; slogp[t] = logp[t]; }
    __syncthreads();

    int n = blockIdx.x * blockDim.x + t;
    if (n < H) {
        float best0 = -3.4e38f, best1 = -3.4e38f;
        int   arg0 = 0, arg1 = 0;
        uint32_t base = (uint32_t)n * (uint32_t)KCOMP;
        #pragma unroll 4
        for (int k = 0; k < KCOMP; ++k) {
            uint32_t i0 = base + (uint32_t)k;
            TF2 bits = threefry2x32_key42(i0, i0 + NK2);
            float lp = slogp[k];
            float s0 = lp + gumbel_from_bits(bits.a);
            float s1 = lp + gumbel_from_bits(bits.b);
            if (s0 > best0) { best0 = s0; arg0 = k; }
            if (s1 > best1) { best1 = s1; arg1 = k; }
        }
        z[n]     = arg0;
        z[n + H] = arg1;
        atomicAdd(&lhist[arg0], 1);
        atomicAdd(&lhist[arg1], 1);
    }
    __syncthreads();
    if (t < KCOMP && lhist[t] != 0) atomicAdd(&hist[t], lhist[t]);
}

// ---------------------------------------------------------------------------
// Kernel 2: exclusive scan of the 64-entry histogram (trivial)
// ---------------------------------------------------------------------------
__global__ void k_scan(const int* __restrict__ hist, int* __restrict__ offs) {
    if (threadIdx.x == 0) {
        int s = 0;
        for (int i = 0; i < KCOMP; ++i) { offs[i] = s; s += hist[i]; }
    }
}

// ---------------------------------------------------------------------------
// Kernel 3: scatter sample ids into per-component bins
// ---------------------------------------------------------------------------
__global__ __launch_bounds__(256)
void k_scatter(const int* __restrict__ z, const int* __restrict__ offs,
               int* __restrict__ fill, int* __restrict__ sortedIdx, int N) {
    int n = blockIdx.x * blockDim.x + threadIdx.x;
    if (n < N) {
        int c = z[n];
        int p = atomicAdd(&fill[c], 1);
        sortedIdx[offs[c] + p] = n;
    }
}

// ---------------------------------------------------------------------------
// Kernel 4: per-component 16-row tiles: D(16x16) = A(16x16) * L_k(16x16) + mu
// via 4 chained V_WMMA_F32_16X16X4_F32 (full fp32, matches reference math).
// grid = (64 components, J column-blocks), 256 threads = 8 wave32.
// Index loads are hoisted ahead of the WMMA chain; the (rare) partial tile
// takes a guarded slow path, full tiles run guard-free.
// ---------------------------------------------------------------------------
__global__ __launch_bounds__(256)
void k_wmma(const float* __restrict__ x, const float* __restrict__ means,
            const float* __restrict__ Ls, const int* __restrict__ sortedIdx,
            const int* __restrict__ offs, const int* __restrict__ hist,
            float* __restrict__ y) {
    const int k    = blockIdx.x;                    // component id
    const int lane = threadIdx.x & 31;
    const int half = lane >> 4;                     // 0: lanes 0-15, 1: 16-31
    const int r    = lane & 15;                     // row (A) / col (B,C,D)

    const int count = hist[k];
    const int base0 = offs[k];
    const int ntile = (count + 15) >> 4;

    // B fragments for the 4 K-slices of L_k (B is K x N = L row-major):
    // VGPR0: K=4j+2*half, VGPR1: K=4j+2*half+1, column = lane&15
    const float* __restrict__ L = Ls + k * (DDIM * DDIM);
    v2f b[4];
    #pragma unroll
    for (int j = 0; j < 4; ++j) {
        b[j].x = L[(4 * j + 2 * half + 0) * DDIM + r];
        b[j].y = L[(4 * j + 2 * half + 1) * DDIM + r];
    }
    const float mu = means[k * DDIM + r];

    const int wid    = blockIdx.y * (blockDim.x >> 5) + (threadIdx.x >> 5);
    const int wcount = gridDim.y * (blockDim.x >> 5);

    for (int t = wid; t < ntile; t += wcount) {
        const int tb   = base0 + t * 16;
        const int m    = min(16, count - t * 16);   // valid rows (wave-uniform)
        const bool full = (m == 16);                // common case: ~99.9%

        // ---- hoisted index loads (independent of WMMA result) -------------
        const int ra   = full ? r : ((r < m) ? r : (m - 1));
        const int idxA = sortedIdx[tb + ra];
        // Output-row indices for this lane's 8 D rows (M = 8*half + v).
        // Unconditional loads are safe: worst case reads a few ints past the
        // bin into adjacent workspace; values are only used for valid rows.
        int idxS[8];
        #pragma unroll
        for (int v = 0; v < 8; ++v)
            idxS[v] = sortedIdx[tb + 8 * half + v];

        // ---- A fragments: row = lane&15, K-slice per half -----------------
        const float* __restrict__ xr = x + (size_t)idxA * DDIM + 2 * half;
        v2f a[4];
        #pragma unroll
        for (int j = 0; j < 4; ++j)
            a[j] = *(const v2f*)(xr + 4 * j);       // K = 4j+2*half, +1

        // ---- 16x16x16 f32 GEMM as 4 chained 16x16x4 WMMAs -----------------
        v8f c = {0.f, 0.f, 0.f, 0.f, 0.f, 0.f, 0.f, 0.f};
        c = __builtin_amdgcn_wmma_f32_16x16x4_f32(false, a[0], false, b[0],
                                                  (short)0, c, false, false);
        c = __builtin_amdgcn_wmma_f32_16x16x4_f32(false, a[1], false, b[1],
                                                  (short)0, c, false, false);
        c = __builtin_amdgcn_wmma_f32_16x16x4_f32(false, a[2], false, b[2],
                                                  (short)0, c, false, false);
        c = __builtin_amdgcn_wmma_f32_16x16x4_f32(false, a[3], false, b[3],
                                                  (short)0, c, false, false);

        // ---- epilogue: +mu broadcast along columns, scatter rows ----------
        // D layout: VGPR v -> M = v + 8*half, N = lane&15. For each store op
        // lanes 0-15 share a row index (contiguous 64B), lanes 16-31 another.
        if (full) {
            #pragma unroll
            for (int v = 0; v < 8; ++v)
                y[(size_t)idxS[v] * DDIM + r] = c[v] + mu;
        } else {
            #pragma unroll
            for (int v = 0; v < 8; ++v) {
                const int M = v + 8 * half;
                if (M < m)
                    y[(size_t)idxS[v] * DDIM + r] = c[v] + mu;
            }
        }
    }
}

// ---------------------------------------------------------------------------
// Host-side launcher
// ---------------------------------------------------------------------------
extern "C" void kernel_launch(void* const* d_in, const int* in_sizes, int n_in,
                              void* d_out, int out_size, void* d_ws, size_t ws_size,
                              hipStream_t stream) {
    const float* x     = (const float*)d_in[0];   // (N, 16)
    const float* pi    = (const float*)d_in[1];   // (64,)
    const float* means = (const float*)d_in[2];   // (64, 16)
    const float* Ls    = (const float*)d_in[3];   // (64, 16, 16)
    float*       y     = (float*)d_out;           // (N, 16)

    const int N  = in_sizes[0] / DDIM;            // 1,000,000
    const int H  = N / 2;                          // N is even
    const uint32_t NK2 = (uint32_t)H * (uint32_t)KCOMP;  // N*K/2 = 32,000,000

    // Workspace layout (all 4-byte typed):
    int*   z         = (int*)d_ws;                // N
    int*   sortedIdx = z + N;                     // N
    int*   hist      = sortedIdx + N;             // 64
    int*   offs      = hist + KCOMP;              // 64
    int*   fill      = offs + KCOMP;              // 64
    float* logp      = (float*)(fill + KCOMP);    // 64

    (void)n_in; (void)out_size; (void)ws_size;

    // 0) zero counters + log-probabilities
    k_init<<<1, KCOMP, 0, stream>>>(pi, hist, fill, logp);

    // 1) Threefry + Gumbel argmax sampling (paired counters -> 2 samples/thread)
    k_sample<<<(H + 255) / 256, 256, 0, stream>>>(logp, z, hist, H, NK2);

    // 2) exclusive scan of 64-bin histogram
    k_scan<<<1, 64, 0, stream>>>(hist, offs);

    // 3) counting-sort scatter of sample indices
    k_scatter<<<(N + 255) / 256, 256, 0, stream>>>(z, offs, fill, sortedIdx, N);

    // 4) WMMA batched matvec per component bin
    dim3 grid4(KCOMP, 32);
    k_wmma<<<grid4, 256, 0, stream>>>(x, means, Ls, sortedIdx, offs, hist, y);
}